// ViewMorphing_71768903516714
// MI455X (gfx1250) — compile-verified
//
#include <hip/hip_runtime.h>

#define IMG_D  224
#define IMG_HW (IMG_D * IMG_D)

typedef float float4v __attribute__((ext_vector_type(4)));

__device__ __forceinline__ float clipq(float v) {
    v = fmaxf(v, 0.001f);
    v = fminf(v, (float)IMG_D - 1.001f);
    return v;
}

// Bilinear 4-corner sample of a 3-channel [3, HW] image plane, reference semantics:
// w = (1-|qx-nx|)(1-|qy-ny|) summed over the 4 floor/ceil corners (double-counts when
// floor==ceil, exactly like the reference).
__device__ __forceinline__ void bilerp3(const float* __restrict__ img,
                                        float qx, float qy,
                                        float& s0, float& s1, float& s2) {
    const float xf = floorf(qx), xc = ceilf(qx);
    const float yf = floorf(qy), yc = ceilf(qy);
    const float wx0 = 1.0f - (qx - xf);
    const float wx1 = 1.0f - (xc - qx);
    const float wy0 = 1.0f - (qy - yf);
    const float wy1 = 1.0f - (yc - qy);
    const int ixf = (int)xf, ixc = (int)xc;
    const int iyf = (int)yf, iyc = (int)yc;
    const int i00 = iyf + IMG_D * ixf;
    const int i01 = iyc + IMG_D * ixf;
    const int i10 = iyf + IMG_D * ixc;
    const int i11 = iyc + IMG_D * ixc;
    const float w00 = wx0 * wy0, w01 = wx0 * wy1;
    const float w10 = wx1 * wy0, w11 = wx1 * wy1;
    const float* __restrict__ c0 = img;
    const float* __restrict__ c1 = img + IMG_HW;
    const float* __restrict__ c2 = img + 2 * IMG_HW;
    s0 = fmaf(w00, c0[i00], fmaf(w01, c0[i01], fmaf(w10, c0[i10], w11 * c0[i11])));
    s1 = fmaf(w00, c1[i00], fmaf(w01, c1[i01], fmaf(w10, c1[i10], w11 * c1[i11])));
    s2 = fmaf(w00, c2[i00], fmaf(w01, c2[i01], fmaf(w10, c2[i10], w11 * c2[i11])));
}

__global__ __launch_bounds__(256) void ViewMorphing_kernel(
    const float* __restrict__ im1, const float* __restrict__ im2,
    const float* __restrict__ C,   const float* __restrict__ M1,
    const float* __restrict__ M2,  float* __restrict__ out)
{
    const int n = blockIdx.y;
    const int t = blockIdx.x * 256 + threadIdx.x;   // 0 .. HW/4 - 1 (exact)
    const int p = t * 4;                            // first of 4 consecutive pixels
    const int row  = p / IMG_D;                     // constant across the 4 (224 % 4 == 0)
    const int col0 = p - row * IMG_D;

    // Streaming operands: non-temporal b128 loads (read exactly once; keep caches
    // free for the image gathers which have high reuse).
    const size_t cbase = (size_t)n * 2 * IMG_HW + p;
    const float4v cx = __builtin_nontemporal_load((const float4v*)(C + cbase));           // row offset
    const float4v cy = __builtin_nontemporal_load((const float4v*)(C + cbase + IMG_HW));  // col offset
    const float4v m1 = __builtin_nontemporal_load((const float4v*)(M1 + (size_t)n * IMG_HW + p));
    const float4v m2 = __builtin_nontemporal_load((const float4v*)(M2 + (size_t)n * IMG_HW + p));

    const float* __restrict__ img1 = im1 + (size_t)n * 3 * IMG_HW;
    const float* __restrict__ img2 = im2 + (size_t)n * 3 * IMG_HW;

    float4v o0, o1, o2;
    const float frow = (float)row;

#pragma unroll
    for (int j = 0; j < 4; ++j) {
        const float fcol = (float)(col0 + j);
        // forward warp: q + C, sample im1
        const float ax = clipq(frow + cx[j]);
        const float ay = clipq(fcol + cy[j]);
        float a0, a1, a2;
        bilerp3(img1, ax, ay, a0, a1, a2);
        // backward warp: q - C, sample im2
        const float bx = clipq(frow - cx[j]);
        const float by = clipq(fcol - cy[j]);
        float b0, b1, b2;
        bilerp3(img2, bx, by, b0, b1, b2);

        const float w1 = m1[j], w2 = m2[j];
        o0[j] = fmaf(w1, a0, w2 * b0);
        o1[j] = fmaf(w1, a1, w2 * b1);
        o2[j] = fmaf(w1, a2, w2 * b2);
    }

    // Streaming output: non-temporal b128 stores, never re-read.
    float* __restrict__ obase = out + (size_t)n * 3 * IMG_HW + p;
    __builtin_nontemporal_store(o0, (float4v*)(obase));
    __builtin_nontemporal_store(o1, (float4v*)(obase + IMG_HW));
    __builtin_nontemporal_store(o2, (float4v*)(obase + 2 * IMG_HW));
}

extern "C" void kernel_launch(void* const* d_in, const int* in_sizes, int n_in,
                              void* d_out, int out_size, void* d_ws, size_t ws_size,
                              hipStream_t stream) {
    const float* im1 = (const float*)d_in[0];
    const float* im2 = (const float*)d_in[1];
    const float* C   = (const float*)d_in[2];
    const float* M1  = (const float*)d_in[3];
    const float* M2  = (const float*)d_in[4];
    float* out = (float*)d_out;

    const int N = in_sizes[0] / (3 * IMG_HW);       // 64
    dim3 grid(IMG_HW / 4 / 256, N);                  // (49, 64), exact coverage
    ViewMorphing_kernel<<<grid, dim3(256, 1, 1), 0, stream>>>(im1, im2, C, M1, M2, out);
}